// GraphSAGENet_34772055228551
// MI455X (gfx1250) — compile-verified
//
#include <hip/hip_runtime.h>
#include <hip/hip_bf16.h>

typedef __attribute__((ext_vector_type(2))) float v2f;
typedef __attribute__((ext_vector_type(8))) float v8f;

#define N_NODES 100000
#define N_EDGES 1600000
#define IN_CH 64
#define HID 16
#define OUT_CH 32
#define L2EPS 1e-12f

// ---------------------------------------------------------------------------
// Zero-fill (accumulators must be re-zeroed deterministically every call)
// ---------------------------------------------------------------------------
__global__ void zero_kernel(float* __restrict__ p, int n) {
    int i = blockIdx.x * blockDim.x + threadIdx.x;
    if (i < n) p[i] = 0.0f;
}

// ---------------------------------------------------------------------------
// GEMM 1: p1 = x @ w1_l^T, r1 = x @ w1_r^T   ([N,64] x [64,16])
// One wave per 16-node tile; V_WMMA_F32_16X16X4_F32, K accumulated 4 at a time.
// A layout: lane L -> row m=L%16, k = 2*(L/16)+{0,1}  (float2)
// B layout: lane L -> col n=L%16, k = 2*(L/16)+{0,1}; B[k][n] = W[n][k]
// D layout: vgpr j -> row j+8*(L/16), col L%16
// ---------------------------------------------------------------------------
__global__ __launch_bounds__(320) void gemm1_kernel(
    const float* __restrict__ x, const float* __restrict__ w_l,
    const float* __restrict__ w_r, float* __restrict__ p1,
    float* __restrict__ r1)
{
    const int wave = blockIdx.x * (blockDim.x >> 5) + (threadIdx.x >> 5);
    const int lane = threadIdx.x & 31;
    const int mn   = lane & 15;            // A row / B col
    const int half = lane >> 4;
    const int kb   = half << 1;            // 0 or 2
    const int node = wave * 16 + mn;

    const float* xrow  = x   + node * IN_CH + kb;
    const float* wlrow = w_l + mn   * IN_CH + kb;   // row n of w1_l
    const float* wrrow = w_r + mn   * IN_CH + kb;

    v8f accL = {}; v8f accR = {};
#pragma unroll
    for (int k0 = 0; k0 < IN_CH; k0 += 4) {
        v2f a  = *(const v2f*)(xrow  + k0);
        v2f bl = *(const v2f*)(wlrow + k0);
        v2f br = *(const v2f*)(wrrow + k0);
        accL = __builtin_amdgcn_wmma_f32_16x16x4_f32(false, a, false, bl,
                                                     (short)0, accL, false, false);
        accR = __builtin_amdgcn_wmma_f32_16x16x4_f32(false, a, false, br,
                                                     (short)0, accR, false, false);
    }
#pragma unroll
    for (int j = 0; j < 8; ++j) {
        int row = wave * 16 + j + 8 * half;
        p1[row * HID + mn] = accL[j];
        r1[row * HID + mn] = accR[j];
    }
}

// ---------------------------------------------------------------------------
// GEMM 2: p2 = h @ w2_l^T, r2 = h @ w2_r^T   ([N,16] x [16,32])
// N=32 -> two 16-wide column tiles per weight.
// ---------------------------------------------------------------------------
__global__ __launch_bounds__(320) void gemm2_kernel(
    const float* __restrict__ h, const float* __restrict__ w_l,
    const float* __restrict__ w_r, float* __restrict__ p2,
    float* __restrict__ r2)
{
    const int wave = blockIdx.x * (blockDim.x >> 5) + (threadIdx.x >> 5);
    const int lane = threadIdx.x & 31;
    const int mn   = lane & 15;
    const int half = lane >> 4;
    const int kb   = half << 1;
    const int node = wave * 16 + mn;

    v8f aL0 = {}, aL1 = {}, aR0 = {}, aR1 = {};
#pragma unroll
    for (int k0 = 0; k0 < HID; k0 += 4) {
        v2f a   = *(const v2f*)(h   + node * HID + k0 + kb);
        v2f bl0 = *(const v2f*)(w_l + (mn     ) * HID + k0 + kb);
        v2f bl1 = *(const v2f*)(w_l + (mn + 16) * HID + k0 + kb);
        v2f br0 = *(const v2f*)(w_r + (mn     ) * HID + k0 + kb);
        v2f br1 = *(const v2f*)(w_r + (mn + 16) * HID + k0 + kb);
        aL0 = __builtin_amdgcn_wmma_f32_16x16x4_f32(false, a, false, bl0,
                                                    (short)0, aL0, false, false);
        aL1 = __builtin_amdgcn_wmma_f32_16x16x4_f32(false, a, false, bl1,
                                                    (short)0, aL1, false, false);
        aR0 = __builtin_amdgcn_wmma_f32_16x16x4_f32(false, a, false, br0,
                                                    (short)0, aR0, false, false);
        aR1 = __builtin_amdgcn_wmma_f32_16x16x4_f32(false, a, false, br1,
                                                    (short)0, aR1, false, false);
    }
#pragma unroll
    for (int j = 0; j < 8; ++j) {
        int row = wave * 16 + j + 8 * half;
        p2[row * OUT_CH + mn]      = aL0[j];
        p2[row * OUT_CH + 16 + mn] = aL1[j];
        r2[row * OUT_CH + mn]      = aR0[j];
        r2[row * OUT_CH + 16 + mn] = aR1[j];
    }
}

// ---------------------------------------------------------------------------
// In-degree counts (one atomic per edge; 400KB target fits in L2)
// ---------------------------------------------------------------------------
__global__ void count_kernel(const long long* __restrict__ ei,
                             float* __restrict__ cnt)
{
    int e = blockIdx.x * blockDim.x + threadIdx.x;
    if (e < N_EDGES) {
        int d = (int)ei[N_EDGES + e];
        atomicAdd(&cnt[d], 1.0f);
    }
}

// ---------------------------------------------------------------------------
// Edge scatter, layer 1: agg1[dst] += p1[src]  (16 lanes per edge -> coalesced)
// ---------------------------------------------------------------------------
__global__ void scatter1_kernel(const long long* __restrict__ ei,
                                const float* __restrict__ p1,
                                float* __restrict__ agg1)
{
    unsigned t = blockIdx.x * blockDim.x + threadIdx.x;
    int e = (int)(t >> 4);
    int c = (int)(t & 15u);
    if (e < N_EDGES) {
        int s = (int)ei[e];
        int d = (int)ei[N_EDGES + e];
        atomicAdd(&agg1[d * HID + c], p1[s * HID + c]);
    }
}

// ---------------------------------------------------------------------------
// Edge scatter, layer 2: agg2[dst] += p2[src]  (32 lanes per edge)
// ---------------------------------------------------------------------------
__global__ void scatter2_kernel(const long long* __restrict__ ei,
                                const float* __restrict__ p2,
                                float* __restrict__ agg2)
{
    unsigned t = blockIdx.x * blockDim.x + threadIdx.x;
    int e = (int)(t >> 5);
    int c = (int)(t & 31u);
    if (e < N_EDGES) {
        int s = (int)ei[e];
        int d = (int)ei[N_EDGES + e];
        atomicAdd(&agg2[d * OUT_CH + c], p2[s * OUT_CH + c]);
    }
}

// ---------------------------------------------------------------------------
// Finalize layer 1: h = relu(l2norm(agg1/max(cnt,1) + b1 + r1))
// 16 lanes per node; row reduction via shfl_xor within 16-lane groups.
// ---------------------------------------------------------------------------
__global__ void finalize1_kernel(const float* __restrict__ agg1,
                                 const float* __restrict__ cnt,
                                 const float* __restrict__ r1,
                                 const float* __restrict__ b1,
                                 float* __restrict__ h)
{
    int t = blockIdx.x * blockDim.x + threadIdx.x;
    int node = t >> 4;
    int c = t & 15;
    float inv = 1.0f / fmaxf(cnt[node], 1.0f);
    float o = agg1[node * HID + c] * inv + b1[c] + r1[node * HID + c];
    float ss = o * o;
    ss += __shfl_xor(ss, 1, 16);
    ss += __shfl_xor(ss, 2, 16);
    ss += __shfl_xor(ss, 4, 16);
    ss += __shfl_xor(ss, 8, 16);
    float nrm = fmaxf(sqrtf(ss), L2EPS);
    h[node * HID + c] = fmaxf(o / nrm, 0.0f);
}

// ---------------------------------------------------------------------------
// Finalize layer 2: out = l2norm(agg2/max(cnt,1) + b2 + r2)
// 32 lanes (full wave) per node.
// ---------------------------------------------------------------------------
__global__ void finalize2_kernel(const float* __restrict__ agg2,
                                 const float* __restrict__ cnt,
                                 const float* __restrict__ r2,
                                 const float* __restrict__ b2,
                                 float* __restrict__ out)
{
    int t = blockIdx.x * blockDim.x + threadIdx.x;
    int node = t >> 5;
    int c = t & 31;
    float inv = 1.0f / fmaxf(cnt[node], 1.0f);
    float o = agg2[node * OUT_CH + c] * inv + b2[c] + r2[node * OUT_CH + c];
    float ss = o * o;
    ss += __shfl_xor(ss, 1, 32);
    ss += __shfl_xor(ss, 2, 32);
    ss += __shfl_xor(ss, 4, 32);
    ss += __shfl_xor(ss, 8, 32);
    ss += __shfl_xor(ss, 16, 32);
    float nrm = fmaxf(sqrtf(ss), L2EPS);
    out[node * OUT_CH + c] = o / nrm;
}

// ---------------------------------------------------------------------------
// Launcher
// ---------------------------------------------------------------------------
extern "C" void kernel_launch(void* const* d_in, const int* in_sizes, int n_in,
                              void* d_out, int out_size, void* d_ws, size_t ws_size,
                              hipStream_t stream) {
    const float*     x    = (const float*)d_in[0];
    const long long* ei   = (const long long*)d_in[1];  // int64 [2, E]
    const float*     w1_l = (const float*)d_in[2];      // [16,64]
    const float*     b1   = (const float*)d_in[3];      // [16]
    const float*     w1_r = (const float*)d_in[4];      // [16,64]
    const float*     w2_l = (const float*)d_in[5];      // [32,16]
    const float*     b2   = (const float*)d_in[6];      // [32]
    const float*     w2_r = (const float*)d_in[7];      // [32,16]
    float* out = (float*)d_out;

    // Workspace layout (floats). agg2 aliases the p1/r1 region (dead by then).
    float* ws   = (float*)d_ws;
    float* p1   = ws;                         // N*16
    float* r1   = ws + (size_t)N_NODES * 16;  // N*16
    float* agg1 = ws + (size_t)N_NODES * 32;  // N*16
    float* cnt  = ws + (size_t)N_NODES * 48;  // N
    float* h    = ws + (size_t)N_NODES * 49;  // N*16
    float* p2   = ws + (size_t)N_NODES * 65;  // N*32
    float* r2   = ws + (size_t)N_NODES * 97;  // N*32  (total N*129 ~ 52MB)
    float* agg2 = ws;                         // N*32, aliases p1+r1

    const int waves_total = N_NODES / 16;     // 6250 (exact)

    // Zero layer-1 accumulators + counts
    zero_kernel<<<(N_NODES * 16 + 255) / 256, 256, 0, stream>>>(agg1, N_NODES * 16);
    zero_kernel<<<(N_NODES + 255) / 256, 256, 0, stream>>>(cnt, N_NODES);

    // Projections for layer 1 (WMMA f32 16x16x4), 10 waves/block
    gemm1_kernel<<<waves_total / 10, 320, 0, stream>>>(x, w1_l, w1_r, p1, r1);

    // Degree counts + layer-1 scatter
    count_kernel<<<(N_EDGES + 255) / 256, 256, 0, stream>>>(ei, cnt);
    scatter1_kernel<<<(N_EDGES * 16 + 255) / 256, 256, 0, stream>>>(ei, p1, agg1);

    // h = relu(l2norm(mean + b1 + lin_r(x)))
    finalize1_kernel<<<(N_NODES * 16 + 255) / 256, 256, 0, stream>>>(agg1, cnt, r1, b1, h);

    // Zero layer-2 accumulator (aliases p1/r1 -- safe only after finalize1)
    zero_kernel<<<(N_NODES * 32 + 255) / 256, 256, 0, stream>>>(agg2, N_NODES * 32);

    // Projections for layer 2 (WMMA), scatter, finalize
    gemm2_kernel<<<waves_total / 10, 320, 0, stream>>>(h, w2_l, w2_r, p2, r2);
    scatter2_kernel<<<(N_EDGES * 32 + 255) / 256, 256, 0, stream>>>(ei, p2, agg2);
    finalize2_kernel<<<(N_NODES * 32 + 255) / 256, 256, 0, stream>>>(agg2, cnt, r2, b2, out);
}